// ScaledDotProductAttention_31885837206187
// MI455X (gfx1250) — compile-verified
//
#include <hip/hip_runtime.h>

typedef __attribute__((ext_vector_type(2))) float v2f;
typedef __attribute__((ext_vector_type(8))) float v8f;

#define N_HEADS 16
#define SEQ     2048
#define DIM     64
#define QTILE   32                 // q rows per workgroup
#define NWAVES  8                  // 256 threads, wave32
#define KSUB    (SEQ / 16)         // 128 key sub-tiles of 16
#define KGROUPS 4                  // key range split across 4 waves per q-subtile
#define KITERS  (KSUB / KGROUPS)   // 32 sub-tiles per wave

// Base-2 softmax: fold 1/sqrt(d_k) and log2(e) into one scale; v_exp_f32 /
// v_log_f32 are natively base-2, accessed RAW via __builtin_amdgcn_exp2f /
// __builtin_amdgcn_logf (no OCML denorm-guard sequence -- softmax terms below
// 2^-126 are zero anyway).
#define SCALE2  (0.125f * 1.44269504088896340736f)   // (1/T) * log2(e)
#define LN2     0.69314718055994530942f

static __device__ __forceinline__ float fast_exp2(float x) {
    return __builtin_amdgcn_exp2f(x);       // v_exp_f32
}
static __device__ __forceinline__ float fast_log2(float x) {
    return __builtin_amdgcn_logf(x);        // v_log_f32
}

__global__ __launch_bounds__(256)
void sdpa_wmma_f32_kernel(const float* __restrict__ Q,
                          const float* __restrict__ K,
                          const float* __restrict__ V,
                          float* __restrict__ Out,       // [SEQ, N_HEADS*DIM]
                          float* __restrict__ Attn,      // [N_HEADS, SEQ, SEQ]
                          float* __restrict__ LogAttn)   // [N_HEADS, SEQ, SEQ]
{
    __shared__ float stats[2][KGROUPS][16][2];     // per-row (max2, sum2) partials
    __shared__ float pbuf[NWAVES][16][17];         // P tile staging (C-layout -> A-layout)
    __shared__ float obuf[2][KGROUPS][16][DIM];    // partial O accumulators

    const int tid  = threadIdx.x;
    const int wave = tid >> 5;
    const int lane = tid & 31;
    const int hi   = lane >> 4;        // half-wave: 0 or 1
    const int ln   = lane & 15;
    const int qs   = wave & 1;         // q sub-tile (16 rows) within 32-row tile
    const int g    = wave >> 1;        // key-range group 0..3

    const int b  = blockIdx.y;
    const int q0 = blockIdx.x * QTILE;
    const int qrow = q0 + 16 * qs + ln;    // A-fragment row for this lane

    const float* Qh = Q + (size_t)b * SEQ * DIM;
    const float* Kh = K + (size_t)b * SEQ * DIM;
    const float* Vh = V + (size_t)b * SEQ * DIM;

    // ---- Preload Q A-fragments: D=64 -> 16 chunks of K=4.
    // f32 A 16x4 layout: lane holds row (lane%16), K = 2*(lane/16)+j in VGPR j.
    v2f qa[16];
    #pragma unroll
    for (int c = 0; c < 16; ++c)
        qa[c] = *(const v2f*)(Qh + (size_t)qrow * DIM + 4 * c + 2 * hi);

    // =======================================================================
    // Pass 1: per-LANE online softmax stats in the base-2 domain. Each lane
    // only tracks the columns it personally holds in the C layout (N = lane%16
    // of each k-subtile): the k-loop is pure VALU (max + 2x v_exp + fma per
    // element, no cross-lane traffic). The 16-lane xor-merge happens ONCE.
    // =======================================================================
    float m8[8], s8[8];
    #pragma unroll
    for (int r = 0; r < 8; ++r) { m8[r] = -3.0e38f; s8[r] = 0.0f; }

    for (int it = 0; it < KITERS; ++it) {
        const int kb = g + KGROUPS * it;
        const int krow = kb * 16 + ln;
        v8f acc = {};
        #pragma unroll
        for (int c = 0; c < 16; ++c) {
            v2f kf = *(const v2f*)(Kh + (size_t)krow * DIM + 4 * c + 2 * hi);
            acc = __builtin_amdgcn_wmma_f32_16x16x4_f32(false, qa[c], false, kf,
                                                        (short)0, acc, false, false);
        }
        #pragma unroll
        for (int r = 0; r < 8; ++r) {
            float sv2 = acc[r] * SCALE2;                  // log2-domain score
            float mnew = fmaxf(m8[r], sv2);
            s8[r] = s8[r] * fast_exp2(m8[r] - mnew) + fast_exp2(sv2 - mnew);
            m8[r] = mnew;
        }
    }

    // one-time xor-merge of (m, s) pairs within each 16-lane half
    #pragma unroll
    for (int r = 0; r < 8; ++r) {
        #pragma unroll
        for (int step = 1; step <= 8; step <<= 1) {
            float om = __shfl_xor(m8[r], step, 32);
            float os = __shfl_xor(s8[r], step, 32);
            float M = fmaxf(m8[r], om);
            s8[r] = s8[r] * fast_exp2(m8[r] - M) + os * fast_exp2(om - M);
            m8[r] = M;
        }
    }

    // publish partial stats (one lane per half-wave carries all 8 rows)
    if (ln == 0) {
        #pragma unroll
        for (int r = 0; r < 8; ++r) {
            stats[qs][g][8 * hi + r][0] = m8[r];
            stats[qs][g][8 * hi + r][1] = s8[r];
        }
    }
    __syncthreads();

    // merge across the 4 key groups; every lane keeps stats for its C-layout rows
    float Mrow[8], Lrow[8];
    #pragma unroll
    for (int r = 0; r < 8; ++r) {
        const int row = 8 * hi + r;
        float M = stats[qs][0][row][0];
        M = fmaxf(M, stats[qs][1][row][0]);
        M = fmaxf(M, stats[qs][2][row][0]);
        M = fmaxf(M, stats[qs][3][row][0]);
        float S = stats[qs][0][row][1] * fast_exp2(stats[qs][0][row][0] - M)
                + stats[qs][1][row][1] * fast_exp2(stats[qs][1][row][0] - M)
                + stats[qs][2][row][1] * fast_exp2(stats[qs][2][row][0] - M)
                + stats[qs][3][row][1] * fast_exp2(stats[qs][3][row][0] - M);
        Mrow[r] = M;
        Lrow[r] = fast_log2(S);          // native v_log_f32
    }

    // =======================================================================
    // Pass 2: recompute S (L2-resident K), emit attn/log_attn with NT stores,
    // accumulate O += P @ V via WMMA (P transposed through per-wave LDS).
    // =======================================================================
    v8f oacc[4] = {};   // 4 x (16x16) output tiles covering D=64

    for (int it = 0; it < KITERS; ++it) {
        const int kb = g + KGROUPS * it;
        const int krow = kb * 16 + ln;
        v8f acc = {};
        #pragma unroll
        for (int c = 0; c < 16; ++c) {
            v2f kf = *(const v2f*)(Kh + (size_t)krow * DIM + 4 * c + 2 * hi);
            acc = __builtin_amdgcn_wmma_f32_16x16x4_f32(false, qa[c], false, kf,
                                                        (short)0, acc, false, false);
        }
        #pragma unroll
        for (int r = 0; r < 8; ++r) {
            const int row = 8 * hi + r;                       // row within subtile
            float lo2 = acc[r] * SCALE2 - Mrow[r] - Lrow[r];  // log2-softmax (pk_fma)
            float p   = fast_exp2(lo2);                       // softmax
            float lo  = lo2 * LN2;                            // natural log_softmax
            size_t gi = ((size_t)b * SEQ + (size_t)(q0 + 16 * qs + row)) * SEQ
                      + (size_t)kb * 16 + ln;
            __builtin_nontemporal_store(p,  Attn + gi);     // streamed, never re-read
            __builtin_nontemporal_store(lo, LogAttn + gi);
            pbuf[wave][row][ln] = p;                        // per-wave staging
        }
        // Same-wave LDS ops are in-order: reads below see the writes above.
        v2f pa[4];
        #pragma unroll
        for (int c = 0; c < 4; ++c) {
            pa[c][0] = pbuf[wave][ln][4 * c + 2 * hi];
            pa[c][1] = pbuf[wave][ln][4 * c + 2 * hi + 1];
        }
        #pragma unroll
        for (int t = 0; t < 4; ++t) {
            #pragma unroll
            for (int c = 0; c < 4; ++c) {
                // B 4x16 layout: lane holds column N=lane%16, K = 2*(lane/16)+j
                const float* vp = Vh + (size_t)(kb * 16 + 4 * c + 2 * hi) * DIM
                                     + 16 * t + ln;
                v2f bf;
                bf[0] = vp[0];
                bf[1] = vp[DIM];
                oacc[t] = __builtin_amdgcn_wmma_f32_16x16x4_f32(false, pa[c], false, bf,
                                                                (short)0, oacc[t],
                                                                false, false);
            }
        }
    }

    // ---- Merge partial O across the 4 key groups, write transposed output ----
    #pragma unroll
    for (int t = 0; t < 4; ++t)
        #pragma unroll
        for (int r = 0; r < 8; ++r)
            obuf[qs][g][8 * hi + r][16 * t + ln] = oacc[t][r];
    __syncthreads();

    // 32x64 tile = 2048 elements; 256 threads x 8, coalesced along d
    #pragma unroll
    for (int i = 0; i < 8; ++i) {
        const int idx = tid + 256 * i;
        const int row = idx >> 6;          // 0..31
        const int d   = idx & 63;
        float v = obuf[row >> 4][0][row & 15][d]
                + obuf[row >> 4][1][row & 15][d]
                + obuf[row >> 4][2][row & 15][d]
                + obuf[row >> 4][3][row & 15][d];
        // output[l, b*D + d] = out[b, l, d]
        Out[(size_t)(q0 + row) * (N_HEADS * DIM) + b * DIM + d] = v;
    }
}

extern "C" void kernel_launch(void* const* d_in, const int* in_sizes, int n_in,
                              void* d_out, int out_size, void* d_ws, size_t ws_size,
                              hipStream_t stream) {
    const float* q = (const float*)d_in[0];
    const float* k = (const float*)d_in[1];
    const float* v = (const float*)d_in[2];

    float* out      = (float*)d_out;                                   // [2048, 1024]
    float* attn     = out + (size_t)SEQ * N_HEADS * DIM;               // [16, 2048, 2048]
    float* log_attn = attn + (size_t)N_HEADS * SEQ * SEQ;              // [16, 2048, 2048]

    dim3 grid(SEQ / QTILE, N_HEADS);
    sdpa_wmma_f32_kernel<<<grid, 256, 0, stream>>>(q, k, v, out, attn, log_attn);
}